// QLIFSpike_74062416052608
// MI455X (gfx1250) — compile-verified
//
#include <hip/hip_runtime.h>

// QLIF spiking recurrence, memory-bound streaming kernel for gfx1250.
// Shapes (fixed by reference): x [B=32, T=10, C=128, H=32, W=32] f32,
// mask [B, T, C] i32, out = spikes [B, T, C, H, W] f32.
//
// Roofline: 2 x 167.8 MB of traffic at 23.3 TB/s => ~14.4 us floor; ~0.75
// flop/byte, so purely HBM-bound. No contraction exists, so WMMA cannot
// apply; the CDNA5 features that matter are the async global->LDS copy
// engine (ASYNCcnt, double-buffered over T), B128 transactions, wave32
// SALU scalarization of the block-uniform mask, and NT cache hints on
// both streams (335 MB streamed once > 192 MB L2).
//
// Strategy: 1 block per (b, c) pair (4096 blocks x 256 threads, 8 wave32s).
// Each thread owns 4 consecutive floats of the 1024-float (h,w) tile.
// Per-timestep tiles are staged global->LDS with global_load_async_to_lds,
// double-buffered: issue t+1 while consuming t, split s_wait_asynccnt.

typedef float v4f __attribute__((ext_vector_type(4)));

#define B_  32
#define T_  10
#define C_  128
#define HW_ 1024   // H*W

__global__ __launch_bounds__(256) void qlif_spike_kernel(
    const float* __restrict__ x,
    const int*   __restrict__ mask,
    float*       __restrict__ out)
{
    __shared__ float buf[2 * HW_];               // two 4 KB tiles (8 KB of 320 KB)

    const int tid = threadIdx.x;                 // 0..255
    const int bc  = blockIdx.x;                  // 0..B_*C_-1
    const int b   = bc >> 7;                     // / C_
    const int c   = bc & (C_ - 1);
    const int e4  = tid << 2;                    // float offset of this lane's quad

    const int base_tc = b * T_ * C_ + c;         // index into mask; +t*C_ per step
    const unsigned long long xbase = (unsigned long long)x;
    const unsigned lds0 = (unsigned)(size_t)&buf[e4];   // wave-relative LDS byte addr

    // Issue async copy of timestep t's 16B quad into LDS buffer (t&1).
    // th:TH_LOAD_NT — streamed-once data, don't let it churn L2 against
    // the NT output stream.
    auto issue_async = [&](int t) {
        const unsigned lds  = lds0 + (unsigned)((t & 1) * (HW_ * 4));
        const unsigned voff = ((unsigned)((base_tc + t * C_) * HW_ + e4)) << 2; // bytes
        asm volatile("global_load_async_to_lds_b128 %0, %1, %2 th:TH_LOAD_NT"
                     :: "v"(lds), "v"(voff), "s"(xbase) : "memory");
    };

    issue_async(0);                              // prologue: stage t = 0

    v4f mem = (v4f){0.0f, 0.0f, 0.0f, 0.0f};

#pragma unroll
    for (int t = 0; t < T_; ++t) {
        if (t + 1 < T_) {
            issue_async(t + 1);                  // stage next tile (pipeline depth 2)
            asm volatile("s_wait_asynccnt 0x1" ::: "memory");  // tile t landed
        } else {
            asm volatile("s_wait_asynccnt 0x0" ::: "memory");  // drain
        }

        const float m = (float)mask[base_tc + t * C_];         // block-uniform -> s_load

        const v4f xv = *(const v4f*)&buf[(t & 1) * HW_ + e4];  // ds_load_b128

        v4f sv;
#pragma unroll
        for (int j = 0; j < 4; ++j) {
            float mm = mem[j] * 0.5f + xv[j] * m;              // leaky integrate (TAU=0.5)
            float s  = (mm - 1.0f > 0.0f) ? 1.0f : 0.0f;       // fire at THRESH=1
            mm = (1.0f - s) * mm;                              // hard reset
            mm = fminf(fmaxf(rintf(mm), -3.0f), 1.0f);         // qmem: round-half-even, clip [-3,1]
            mem[j] = mm;
            sv[j]  = s;
        }

        // Streaming output: NT store, 16B per lane, fully coalesced.
        float* op = out + ((size_t)(base_tc + t * C_) * HW_ + e4);
        __builtin_nontemporal_store(sv, (v4f*)op);
    }
}

extern "C" void kernel_launch(void* const* d_in, const int* in_sizes, int n_in,
                              void* d_out, int out_size, void* d_ws, size_t ws_size,
                              hipStream_t stream) {
    (void)in_sizes; (void)n_in; (void)d_ws; (void)ws_size; (void)out_size;

    const float* x    = (const float*)d_in[0];
    const int*   mask = (const int*)d_in[1];
    // d_in[2] (conv_input) is unused by the reference.
    float* out = (float*)d_out;

    dim3 grid(B_ * C_);   // 4096 blocks
    dim3 block(256);      // 8 wave32s
    qlif_spike_kernel<<<grid, block, 0, stream>>>(x, mask, out);
}